// GCN_DGL_58110907515587
// MI455X (gfx1250) — compile-verified
//
#include <hip/hip_runtime.h>
#include <hip/hip_bf16.h>
#include <math.h>

// ---------------------------------------------------------------------------
// 2-layer GCN (DGL GraphConv, norm='both') for MI455X / gfx1250.
// GEMMs use V_WMMA_F32_16X16X4_F32 (fp32 matrix path, full precision);
// edge aggregation uses native global_atomic_add_f32 (memory-bound part).
// ---------------------------------------------------------------------------

typedef __attribute__((ext_vector_type(2))) float v2f;
typedef __attribute__((ext_vector_type(8))) float v8f;

#define N_NODES   100000
#define N_EDGES   1600000
#define IN_FEATS  128
#define N_HIDDEN  64
#define N_CLASSES 40

// ----------------------------- utility kernels -----------------------------

__global__ void zero_f32(float* __restrict__ p, int n) {
  int i = blockIdx.x * blockDim.x + threadIdx.x;
  if (i < n) p[i] = 0.0f;
}

__global__ void degree_kernel(const int* __restrict__ src, const int* __restrict__ dst,
                              float* __restrict__ deg_out, float* __restrict__ deg_in) {
  int e = blockIdx.x * blockDim.x + threadIdx.x;
  if (e >= N_EDGES) return;
  unsafeAtomicAdd(&deg_out[src[e]], 1.0f);
  unsafeAtomicAdd(&deg_in[dst[e]], 1.0f);
}

__global__ void norm_kernel(const float* __restrict__ deg_out, const float* __restrict__ deg_in,
                            float* __restrict__ ns, float* __restrict__ nd) {
  int i = blockIdx.x * blockDim.x + threadIdx.x;
  if (i >= N_NODES) return;
  float o = deg_out[i], d = deg_in[i];
  ns[i] = (o > 0.0f) ? rsqrtf(fmaxf(o, 1.0f)) : 0.0f;
  nd[i] = (d > 0.0f) ? rsqrtf(fmaxf(d, 1.0f)) : 0.0f;
}

// ------------------------- layer 1 GEMM via WMMA ---------------------------
// H = (X * norm_src[:,None]) @ W1   : [100000,128] x [128,64] -> [100000,64]
// One wave computes one 16x16 output tile; K loop over 128 in steps of 4.
__global__ __launch_bounds__(256) void gemm1_wmma(
    const float* __restrict__ X, const float* __restrict__ W,
    const float* __restrict__ norm_src, float* __restrict__ out) {
  const int waves_per_block = 256 / 32;
  int wave = blockIdx.x * waves_per_block + (threadIdx.x >> 5);
  int lane = threadIdx.x & 31;
  int tileN = wave & 3;   // 64/16 = 4 col tiles
  int tileM = wave >> 2;  // 100000/16 = 6250 row tiles (grid sized exactly)
  int half = lane >> 4;   // 0: K+0/1, 1: K+2/3 (A); same split for B rows
  int idx  = lane & 15;

  int row = tileM * 16 + idx;
  float ns = norm_src[row];
  const float* __restrict__ xrow = X + row * IN_FEATS;
  int col = tileN * 16 + idx;

  v8f c = {};
  for (int k0 = 0; k0 < IN_FEATS; k0 += 4) {
    int kk = k0 + 2 * half;
    v2f a, b;
    a.x = xrow[kk]     * ns;
    a.y = xrow[kk + 1] * ns;
    b.x = W[kk * N_HIDDEN + col];
    b.y = W[(kk + 1) * N_HIDDEN + col];
    c = __builtin_amdgcn_wmma_f32_16x16x4_f32(
        /*neg_a=*/false, a, /*neg_b=*/false, b,
        /*c_mod=*/(short)0, c, /*reuse_a=*/false, /*reuse_b=*/false);
  }
#pragma unroll
  for (int r = 0; r < 8; ++r) {
    int m = tileM * 16 + r + 8 * half;
    out[m * N_HIDDEN + col] = c[r];
  }
}

// --------------------------- edge scatter-add ------------------------------
// agg[dst[e], f] += h[src[e], f]   (F = feature width, compile-time)
template <int F>
__global__ void scatter_add(const float* __restrict__ h, const int* __restrict__ src,
                            const int* __restrict__ dst, float* __restrict__ agg) {
  int tid = blockIdx.x * blockDim.x + threadIdx.x;
  if (tid >= N_EDGES * F) return;
  int e = tid / F;
  int f = tid % F;
  int s = src[e];
  int d = dst[e];
  unsafeAtomicAdd(&agg[d * F + f], h[s * F + f]);
}

// ----------------- fused scale + bias + ReLU + next-layer scale ------------
// h1s[i,j] = relu(agg[i,j]*norm_dst[i] + b1[j]) * norm_src[i]
__global__ void act_kernel(const float* __restrict__ agg, const float* __restrict__ nd,
                           const float* __restrict__ ns, const float* __restrict__ b1,
                           float* __restrict__ out) {
  int tid = blockIdx.x * blockDim.x + threadIdx.x;
  if (tid >= N_NODES * N_HIDDEN) return;
  int i = tid >> 6;  // / 64
  int j = tid & 63;
  float v = agg[tid] * nd[i] + b1[j];
  v = fmaxf(v, 0.0f);
  out[tid] = v * ns[i];
}

// ------------------------- layer 2 GEMM via WMMA ---------------------------
// H2 = H1s @ W2 : [100000,64] x [64,40] -> [100000,40]; N padded to 48 (3 tiles),
// columns >= 40 masked at B-load and store (EXEC is full at every WMMA).
__global__ __launch_bounds__(256) void gemm2_wmma(
    const float* __restrict__ H, const float* __restrict__ W2,
    float* __restrict__ out) {
  const int waves_per_block = 256 / 32;
  int wave = blockIdx.x * waves_per_block + (threadIdx.x >> 5);
  int lane = threadIdx.x & 31;
  int tileN = wave % 3;   // 3 col tiles (48 cols, mask >= 40)
  int tileM = wave / 3;
  if (tileM >= N_NODES / 16) return;  // wave-uniform exit
  int half = lane >> 4;
  int idx  = lane & 15;

  int row = tileM * 16 + idx;
  int col = tileN * 16 + idx;
  bool colOK = (col < N_CLASSES);
  const float* __restrict__ hrow = H + row * N_HIDDEN;

  v8f c = {};
  for (int k0 = 0; k0 < N_HIDDEN; k0 += 4) {
    int kk = k0 + 2 * half;
    v2f a, b;
    a.x = hrow[kk];
    a.y = hrow[kk + 1];
    b.x = colOK ? W2[kk * N_CLASSES + col] : 0.0f;
    b.y = colOK ? W2[(kk + 1) * N_CLASSES + col] : 0.0f;
    c = __builtin_amdgcn_wmma_f32_16x16x4_f32(
        false, a, false, b, (short)0, c, false, false);
  }
  if (colOK) {
#pragma unroll
    for (int r = 0; r < 8; ++r) {
      int m = tileM * 16 + r + 8 * half;
      out[m * N_CLASSES + col] = c[r];
    }
  }
}

// --------------------- final scale + bias + log-softmax --------------------
// In-place on d_out: y = out[i,:]*norm_dst[i] + b2; out = y - max - log(sum(exp))
__global__ void softmax_kernel(float* __restrict__ out, const float* __restrict__ nd,
                               const float* __restrict__ b2) {
  int i = blockIdx.x * blockDim.x + threadIdx.x;
  if (i >= N_NODES) return;
  float y[N_CLASSES];
  float nrm = nd[i];
  float m = -INFINITY;
#pragma unroll
  for (int j = 0; j < N_CLASSES; ++j) {
    y[j] = out[i * N_CLASSES + j] * nrm + b2[j];
    m = fmaxf(m, y[j]);
  }
  float s = 0.0f;
#pragma unroll
  for (int j = 0; j < N_CLASSES; ++j) s += expf(y[j] - m);
  float lse = logf(s);
#pragma unroll
  for (int j = 0; j < N_CLASSES; ++j) out[i * N_CLASSES + j] = y[j] - m - lse;
}

// ------------------------------- launcher ----------------------------------

extern "C" void kernel_launch(void* const* d_in, const int* in_sizes, int n_in,
                              void* d_out, int out_size, void* d_ws, size_t ws_size,
                              hipStream_t stream) {
  const float* features = (const float*)d_in[0];  // [100000,128]
  const int*   src      = (const int*)d_in[1];    // [1.6M]
  const int*   dst      = (const int*)d_in[2];    // [1.6M]
  const float* W1       = (const float*)d_in[3];  // [128,64]
  const float* b1       = (const float*)d_in[4];  // [64]
  const float* W2       = (const float*)d_in[5];  // [64,40]
  const float* b2       = (const float*)d_in[6];  // [40]
  float*       outp     = (float*)d_out;          // [100000,40]

  // Workspace layout (floats): deg_out | deg_in | norm_src | norm_dst | h | agg1
  float* ws      = (float*)d_ws;
  float* deg_out = ws;                       // 100000
  float* deg_in  = ws + N_NODES;             // 100000
  float* nsrc    = ws + 2 * N_NODES;         // 100000
  float* ndst    = ws + 3 * N_NODES;         // 100000
  float* h1      = ws + 4 * N_NODES;         // 100000*64 (GEMM1 out, later h1s)
  float* agg1    = h1 + N_NODES * N_HIDDEN;  // 100000*64 (later reused as GEMM2 out)
  float* h2      = agg1;                     // [100000,40] reuse (agg1 dead by then)

  const int B = 256;

  // 1) zero accumulators (deg_out+deg_in contiguous), agg1, and d_out
  zero_f32<<<(2 * N_NODES + B - 1) / B, B, 0, stream>>>(deg_out, 2 * N_NODES);
  zero_f32<<<(N_NODES * N_HIDDEN + B - 1) / B, B, 0, stream>>>(agg1, N_NODES * N_HIDDEN);
  zero_f32<<<(N_NODES * N_CLASSES + B - 1) / B, B, 0, stream>>>(outp, N_NODES * N_CLASSES);

  // 2) degrees + norms
  degree_kernel<<<(N_EDGES + B - 1) / B, B, 0, stream>>>(src, dst, deg_out, deg_in);
  norm_kernel<<<(N_NODES + B - 1) / B, B, 0, stream>>>(deg_out, deg_in, nsrc, ndst);

  // 3) layer 1 GEMM (WMMA): 6250*4 = 25000 tiles, 8 waves/block
  gemm1_wmma<<<25000 / 8, B, 0, stream>>>(features, W1, nsrc, h1);

  // 4) layer 1 edge aggregation
  scatter_add<N_HIDDEN><<<(N_EDGES * N_HIDDEN + B - 1) / B, B, 0, stream>>>(h1, src, dst, agg1);

  // 5) scale + bias + ReLU + pre-scale for layer 2 (h1 buffer reused in place)
  act_kernel<<<(N_NODES * N_HIDDEN + B - 1) / B, B, 0, stream>>>(agg1, ndst, nsrc, b1, h1);

  // 6) layer 2 GEMM (WMMA): 6250*3 = 18750 tiles
  gemm2_wmma<<<(18750 + 7) / 8, B, 0, stream>>>(h1, W2, h2);

  // 7) layer 2 edge aggregation directly into d_out (zeroed above)
  scatter_add<N_CLASSES><<<(N_EDGES * N_CLASSES + B - 1) / B, B, 0, stream>>>(h2, src, dst, outp);

  // 8) scale + bias + log-softmax, in place on d_out
  softmax_kernel<<<(N_NODES + B - 1) / B, B, 0, stream>>>(outp, ndst, b2);
}